// ResidualRotationWarpLayer_5205500362859
// MI455X (gfx1250) — compile-verified
//
#include <hip/hip_runtime.h>
#include <cstdint>
#include <cstddef>

// ---------------------------------------------------------------------------
// Homography warp + bilinear resample, MI455X (gfx1250).
//
// Roofline: ~201 MB moved / 23.3 TB/s ~= 8.7 us floor; ~0.5 GFLOP -> pure
// bandwidth problem, no WMMA role. Strategy: async global->LDS staging of the
// per-tile source window (b128 transfers, ASYNCcnt), LDS gathers, b128 stores.
// ---------------------------------------------------------------------------

namespace {
constexpr int kB = 32, kH = 512, kW = 512, kC = 3;
constexpr int kTile = 32;          // 32x32 output pixels per 256-thread block
constexpr int kPxT  = 4;           // pixels per thread along x (48B = 3x b128 store)
constexpr int kSrcMax = 40;        // max staged source window
constexpr int kLPitch = ((kSrcMax * kC + 3) & ~3); // padded LDS row pitch = 120 floats
constexpr float kOrgX = 256.0f;    // W/2
constexpr float kOrgY = 256.0f;    // H/2
}

// Build per-batch homography H[b] = diag(fx,fy,1) * R * diag(1/fx,1/fy,1).
__global__ void hwarp_prep(const float* __restrict__ focal,
                           const float* __restrict__ q,
                           float* __restrict__ hmat) {
  const int b = threadIdx.x;
  if (b >= kB) return;
  float q0 = q[0], q1 = q[1], q2 = q[2], q3 = q[3];
  const float s = sqrtf(2.0f / (q0 * q0 + q1 * q1 + q2 * q2 + q3 * q3));
  q0 *= s; q1 *= s; q2 *= s; q3 *= s;
  const float Q11 = q1 * q1, Q22 = q2 * q2, Q33 = q3 * q3;
  const float Q12 = q1 * q2, Q13 = q1 * q3, Q23 = q2 * q3;
  const float Q10 = q1 * q0, Q20 = q2 * q0, Q30 = q3 * q0;
  const float R00 = 1.0f - Q22 - Q33, R01 = Q12 - Q30,        R02 = Q13 + Q20;
  const float R10 = Q12 + Q30,        R11 = 1.0f - Q11 - Q33, R12 = Q23 - Q10;
  const float R20 = Q13 - Q20,        R21 = Q23 + Q10,        R22 = 1.0f - Q11 - Q22;
  const float fx = focal[2 * b + 0], fy = focal[2 * b + 1];
  float* h = hmat + b * 12;
  h[0] = R00;           h[1] = R01 * fx / fy; h[2] = R02 * fx;
  h[3] = R10 * fy / fx; h[4] = R11;           h[5] = R12 * fy;
  h[6] = R20 / fx;      h[7] = R21 / fy;      h[8] = R22;
}

__global__ __launch_bounds__(256)
void hwarp_main(const float* __restrict__ img,
                const float* __restrict__ hmat,
                float* __restrict__ out) {
  const int b   = blockIdx.z;
  const int tx0 = blockIdx.x * kTile;
  const int ty0 = blockIdx.y * kTile;
  const int tid = threadIdx.x;
  const int lx4 = (tid & 7) * kPxT;   // 8 threads x 4 px cover a 32-px row
  const int ly  = tid >> 3;           // 32 rows

  // H[b] is block-uniform -> scalarized loads.
  const float* __restrict__ hp = hmat + b * 12;
  const float h00 = hp[0], h01 = hp[1], h02 = hp[2];
  const float h10 = hp[3], h11 = hp[4], h12 = hp[5];
  const float h20 = hp[6], h21 = hp[7], h22 = hp[8];

  __shared__ __align__(16) float tile[kSrcMax * kLPitch]; // 19.2 KB

  // ---- Block-uniform source bbox from warped tile corners.
  // w(x,y) = h20*x + h21*y + h22 is linear: consistent sign at corners =>
  // consistent inside, and warped tile lies in the hull of warped corners.
  float minx = 3.0e38f, maxx = -3.0e38f;
  float miny = 3.0e38f, maxy = -3.0e38f;
  int allpos = 1, allneg = 1;
#pragma unroll
  for (int c = 0; c < 4; ++c) {
    const float cx = (float)(tx0 + (c & 1) * (kTile - 1)) - kOrgX;
    const float cy = (float)(ty0 + ((c >> 1) & 1) * (kTile - 1)) - kOrgY;
    const float wd = h20 * cx + h21 * cy + h22;
    allpos &= (wd > 1e-6f);
    allneg &= (wd < -1e-6f);
    const float iw = 1.0f / wd;
    const float xw = (h00 * cx + h01 * cy + h02) * iw + kOrgX;
    const float yw = (h10 * cx + h11 * cy + h12) * iw + kOrgY;
    minx = fminf(minx, xw); maxx = fmaxf(maxx, xw);
    miny = fminf(miny, yw); maxy = fmaxf(maxy, yw);
  }
  const int signok = allpos | allneg;
  int sx0 = (int)floorf(minx), sx1 = (int)floorf(maxx) + 1; // bilinear footprint
  int sy0 = (int)floorf(miny), sy1 = (int)floorf(maxy) + 1;
  sx0 = max(sx0, 0); sx1 = min(sx1, kW - 1);
  sy0 = max(sy0, 0); sy1 = min(sy1, kH - 1);
  const int rw = sx1 - sx0 + 1;
  const int rh = sy1 - sy0 + 1;

  const int pxb = tx0 + lx4;
  const int py  = ty0 + ly;
  // pxb % 4 == 0 -> pixel index % 4 == 0 -> byte offset % 48 == 0 -> 16B aligned.
  float* __restrict__ op = out + (((size_t)b * kH + py) * kW + pxb) * kC;
  float4* __restrict__ o4 = reinterpret_cast<float4*>(op);

  // Tile entirely off-image: all taps invalid -> zeros (block-uniform).
  if (signok && (rw <= 0 || rh <= 0)) {
    const float4 z = make_float4(0.f, 0.f, 0.f, 0.f);
    o4[0] = z; o4[1] = z; o4[2] = z;
    return;
  }

  const bool use_lds =
      signok && rw > 0 && rh > 0 && rw <= kSrcMax && rh <= kSrcMax;

  if (use_lds) {
    // ---- Async-stage the source window into LDS (gfx1250 ASYNC path).
    // LDS rows padded to 16B multiples so b128 LDS writes are aligned;
    // global side is dword-aligned (legal for multi-dword global ops).
    const int rowlen = rw * kC;                 // floats per source row
    const int lpitch = (rowlen + 3) & ~3;       // padded LDS pitch (floats)
    const int cpr    = lpitch >> 2;             // b128 chunks per row
    const int nch    = rh * cpr;
    const unsigned lbase = (unsigned)(uintptr_t)(&tile[0]);
    const float* __restrict__ gb = img + (size_t)b * kH * kW * kC;
    for (int idx = tid; idx < nch; idx += 256) {
      const int r   = idx / cpr;
      const int off = (idx - r * cpr) * 4;
      const float* grow = gb + ((size_t)(sy0 + r) * kW + sx0) * kC;
      if (off + 4 <= rowlen) {
        const unsigned laddr = lbase + 4u * (unsigned)(r * lpitch + off);
        const uint64_t ga = (uint64_t)(uintptr_t)(grow + off);
        asm volatile("global_load_async_to_lds_b128 %0, %1, off"
                     :: "v"(laddr), "v"(ga) : "memory");
      } else {
        for (int k = off; k < rowlen; ++k) {    // 1..3 tail floats
          const unsigned la = lbase + 4u * (unsigned)(r * lpitch + k);
          const uint64_t ga = (uint64_t)(uintptr_t)(grow + k);
          asm volatile("global_load_async_to_lds_b32 %0, %1, off"
                       :: "v"(la), "v"(ga) : "memory");
        }
      }
    }
    asm volatile("s_wait_asynccnt 0" ::: "memory");
    __syncthreads();
  }

  // ---- 4 output pixels per thread.
  float res[kPxT * kC];
  const float fpy = (float)py - kOrgY;
  const int lpitch = (rw * kC + 3) & ~3;
#pragma unroll
  for (int k = 0; k < kPxT; ++k) {
    const int px = pxb + k;
    const float fpx = (float)px - kOrgX;
    const float wd = h20 * fpx + h21 * fpy + h22;
    const float iw = 1.0f / wd;
    const float xw = (h00 * fpx + h01 * fpy + h02) * iw + kOrgX;
    const float yw = (h10 * fpx + h11 * fpy + h12) * iw + kOrgY;
    const float x0f = floorf(xw), y0f = floorf(yw);
    const float wx = xw - x0f, wy = yw - y0f;
    const int x0i = (int)x0f, y0i = (int)y0f;
    const float w00 = (1.0f - wx) * (1.0f - wy);
    const float w10 = wx * (1.0f - wy);
    const float w01 = (1.0f - wx) * wy;
    const float w11 = wx * wy;

    float a0 = 0.f, a1 = 0.f, a2 = 0.f;
    if (use_lds) {
#pragma unroll
      for (int c = 0; c < 4; ++c) {
        const int xi = x0i + (c & 1);
        const int yi = y0i + (c >> 1);
        const float wgt = (c == 0) ? w00 : (c == 1) ? w10 : (c == 2) ? w01 : w11;
        const int v = (xi >= 0) & (xi < kW) & (yi >= 0) & (yi < kH);
        const float wv = v ? wgt : 0.0f;
        // In-image taps are inside [sx0,sx1]x[sy0,sy1] by construction.
        const int cx = min(max(xi - sx0, 0), rw - 1);
        const int cy = min(max(yi - sy0, 0), rh - 1);
        const float* p = &tile[cy * lpitch + cx * kC];
        a0 += wv * p[0]; a1 += wv * p[1]; a2 += wv * p[2];
      }
    } else {
      // Generic fallback: clamped global gathers (always correct).
#pragma unroll
      for (int c = 0; c < 4; ++c) {
        const int xi = x0i + (c & 1);
        const int yi = y0i + (c >> 1);
        const float wgt = (c == 0) ? w00 : (c == 1) ? w10 : (c == 2) ? w01 : w11;
        const int v = (xi >= 0) & (xi < kW) & (yi >= 0) & (yi < kH);
        const float wv = v ? wgt : 0.0f;
        const int cx = min(max(xi, 0), kW - 1);
        const int cy = min(max(yi, 0), kH - 1);
        const float* p = img + (((size_t)b * kH + cy) * kW + cx) * kC;
        a0 += wv * p[0]; a1 += wv * p[1]; a2 += wv * p[2];
      }
    }
    res[k * kC + 0] = a0; res[k * kC + 1] = a1; res[k * kC + 2] = a2;
  }

  // 48 contiguous, 16B-aligned bytes -> 3x global_store_b128.
  o4[0] = make_float4(res[0], res[1], res[2],  res[3]);
  o4[1] = make_float4(res[4], res[5], res[6],  res[7]);
  o4[2] = make_float4(res[8], res[9], res[10], res[11]);
}

extern "C" void kernel_launch(void* const* d_in, const int* in_sizes, int n_in,
                              void* d_out, int out_size, void* d_ws, size_t ws_size,
                              hipStream_t stream) {
  (void)in_sizes; (void)n_in; (void)out_size; (void)ws_size;
  const float* images = (const float*)d_in[0]; // [32,512,512,3] f32
  const float* focal  = (const float*)d_in[1]; // [32,2] f32
  const float* q      = (const float*)d_in[2]; // [4] f32
  float* out  = (float*)d_out;                 // [32,512,512,3] f32
  float* hmat = (float*)d_ws;                  // 32 * 12 floats scratch

  hwarp_prep<<<1, 32, 0, stream>>>(focal, q, hmat);

  dim3 grid(kW / kTile, kH / kTile, kB);       // 16 x 16 x 32 = 8192 blocks
  hwarp_main<<<grid, 256, 0, stream>>>(images, hmat, out);
}